// DGCNN_PAConv_37933151158908
// MI455X (gfx1250) — compile-verified
//
#include <hip/hip_runtime.h>
#include <math.h>

// ---- problem constants ----
#define Bc 8
#define Nc 1024
#define Cc 3
#define Kc 20
#define Mc 8
#define Oc 64
#define Hc 16
#define EPSc 1e-5f
#define BIASc 0.5f

typedef __attribute__((ext_vector_type(2))) float v2f;
typedef __attribute__((ext_vector_type(8))) float v8f;

// ---------------------------------------------------------------------------
// Kernel 1: kNN.  One workgroup = (batch b, 16-row tile).  16 waves.
// Gram matrix tiles via V_WMMA_F32_16X16X4_F32 (K=3 padded to 4), ranking
// value = 2*inner(i,j) - ||xj||^2  (dropping the row-constant -||xi||^2).
// Per-lane register top-20 insertion, then 20-round cross-lane argmax merge.
// ---------------------------------------------------------------------------
__global__ __launch_bounds__(512) void knn_kernel(const float* __restrict__ coords,
                                                  int* __restrict__ idx_out) {
  const int lane = threadIdx.x & 31;
  const int w    = threadIdx.x >> 5;        // wave id 0..15 (row within tile)
  const int b    = blockIdx.x >> 6;         // 64 row-tiles per batch
  const int r0   = (blockIdx.x & 63) << 4;
  const float* cb = coords + (size_t)b * Cc * Nc;

  __shared__ float lneg[16][260];           // 16 rows x 256-col chunk (+pad)

  // A fragment: rows r0..r0+15, dims 0..2 (K=3 padded with 0 at K=3)
  v2f afrag;
  {
    int m = lane & 15;
    if (lane < 16) { afrag.x = cb[0 * Nc + r0 + m]; afrag.y = cb[1 * Nc + r0 + m]; }
    else           { afrag.x = cb[2 * Nc + r0 + m]; afrag.y = 0.0f; }
  }

  float val[20]; int vidx[20];
#pragma unroll
  for (int j = 0; j < 20; ++j) { val[j] = -__builtin_inff(); vidx[j] = -1; }

  for (int cc = 0; cc < 4; ++cc) {          // 4 chunks of 256 columns
    const int c0 = cc << 8;
    {
      const int nloc = lane & 15;
      const int col  = c0 + (w << 4) + nloc;
      v2f bfrag;
      if (lane < 16) { bfrag.x = cb[0 * Nc + col]; bfrag.y = cb[1 * Nc + col]; }
      else           { bfrag.x = cb[2 * Nc + col]; bfrag.y = 0.0f; }
      v8f c = {};
      c = __builtin_amdgcn_wmma_f32_16x16x4_f32(false, afrag, false, bfrag,
                                                (short)0, c, false, false);
      const float x0 = cb[0 * Nc + col], x1 = cb[1 * Nc + col], x2 = cb[2 * Nc + col];
      const float xx = x0 * x0 + x1 * x1 + x2 * x2;
      const int mbase = (lane >> 4) << 3;
#pragma unroll
      for (int r = 0; r < 8; ++r)
        lneg[mbase + r][(w << 4) + nloc] = 2.0f * c[r] - xx;
    }
    __syncthreads();
    // scan phase: wave w scans its row w over this 256-col chunk
#pragma unroll
    for (int t = 0; t < 8; ++t) {
      const int c = (t << 5) + lane;
      float v = lneg[w][c];
      int  gi = c0 + c;
#pragma unroll
      for (int j = 0; j < 20; ++j) {        // sorted-descending push-down insert
        const bool takes = v > val[j];
        const float tv = takes ? val[j]  : v;
        const int   ti = takes ? vidx[j] : gi;
        val[j]  = takes ? v  : val[j];
        vidx[j] = takes ? gi : vidx[j];
        v = tv; gi = ti;
      }
    }
    __syncthreads();
  }

  // merge: 20 rounds of cross-lane argmax (val desc, index asc tie-break)
  const int n = r0 + w;
  int* orow = idx_out + ((size_t)(b * Nc + n)) * Kc;
#pragma unroll 1
  for (int t = 0; t < 20; ++t) {
    float mv = val[0]; int mi = vidx[0]; int ml = lane;
#pragma unroll
    for (int s = 16; s > 0; s >>= 1) {
      const float ov = __shfl_xor(mv, s, 32);
      const int   oi = __shfl_xor(mi, s, 32);
      const int   ol = __shfl_xor(ml, s, 32);
      const bool take = (ov > mv) || ((ov == mv) && (oi < mi));
      mv = take ? ov : mv; mi = take ? oi : mi; ml = take ? ol : ml;
    }
    if (lane == 0) orow[t] = mi;
    if (lane == ml) {                       // winner shifts its list down
#pragma unroll
      for (int j = 0; j < 19; ++j) { val[j] = val[j + 1]; vidx[j] = vidx[j + 1]; }
      val[19] = -__builtin_inff(); vidx[19] = -1;
    }
  }
}

// ---------------------------------------------------------------------------
// Kernel 2: point[b,n,m,o] = f[b,n,:] @ (matrice1[0:3]+matrice1[3:6])
// GEMM [16 x 3] @ [3 x 512] per tile via V_WMMA_F32_16X16X4_F32.
// ---------------------------------------------------------------------------
__global__ __launch_bounds__(256) void point_kernel(const float* __restrict__ features,
                                                    const float* __restrict__ matrice1,
                                                    float* __restrict__ point) {
  const int lane = threadIdx.x & 31;
  const int w    = threadIdx.x >> 5;        // 0..7
  const int b    = blockIdx.x >> 6;
  const int n0   = (blockIdx.x & 63) << 4;
  const float* fb = features + (size_t)b * Cc * Nc;
  const int m = lane & 15;

  v2f afrag;
  if (lane < 16) { afrag.x = fb[0 * Nc + n0 + m]; afrag.y = fb[1 * Nc + n0 + m]; }
  else           { afrag.x = fb[2 * Nc + n0 + m]; afrag.y = 0.0f; }

  for (int j = 0; j < 4; ++j) {
    const int col = (((w << 2) + j) << 4) + m;   // 0..511
    v2f bfrag;
    if (lane < 16) {
      bfrag.x = matrice1[0 * 512 + col] + matrice1[3 * 512 + col];
      bfrag.y = matrice1[1 * 512 + col] + matrice1[4 * 512 + col];
    } else {
      bfrag.x = matrice1[2 * 512 + col] + matrice1[5 * 512 + col];
      bfrag.y = 0.0f;
    }
    v8f c = {};
    c = __builtin_amdgcn_wmma_f32_16x16x4_f32(false, afrag, false, bfrag,
                                              (short)0, c, false, false);
    const int mb = (lane >> 4) << 3;
#pragma unroll
    for (int r = 0; r < 8; ++r)
      point[((size_t)(b * Nc + n0 + mb + r)) * 512 + col] = c[r];
  }
}

// ---------------------------------------------------------------------------
// ScoreNet helpers + kernels (deterministic two-stage BN stats)
// ---------------------------------------------------------------------------
__device__ inline void sn_hidden(const float* __restrict__ cb, int n, int j,
                                 const float* __restrict__ w1, float* h) {
  const float p0 = cb[n], p1 = cb[Nc + n], p2 = cb[2 * Nc + n];
  const float q0 = cb[j], q1 = cb[Nc + j], q2 = cb[2 * Nc + j];
  const float xyz[6] = {q0 - p0, q1 - p1, q2 - p2, q0, q1, q2};
#pragma unroll
  for (int c = 0; c < Hc; ++c) {
    float a = 0.f;
#pragma unroll
    for (int t = 0; t < 6; ++t) a += xyz[t] * w1[c * 6 + t];
    h[c] = a;
  }
}

__global__ __launch_bounds__(256) void sn_partial_kernel(const float* __restrict__ coords,
                                                         const int* __restrict__ idx,
                                                         const float* __restrict__ sn_w1,
                                                         float* __restrict__ partials) {
  const int gid = blockIdx.x * 256 + threadIdx.x;          // < 163840 exactly
  const int b = gid / (Nc * Kc);
  const int rem = gid % (Nc * Kc);
  const int n = rem / Kc;
  const int j = idx[gid];
  float h[Hc];
  sn_hidden(coords + (size_t)b * Cc * Nc, n, j, sn_w1, h);

  __shared__ float red[256];
  for (int c = 0; c < Hc; ++c) {
    red[threadIdx.x] = h[c];
    __syncthreads();
    for (int st = 128; st > 0; st >>= 1) {
      if (threadIdx.x < st) red[threadIdx.x] += red[threadIdx.x + st];
      __syncthreads();
    }
    if (threadIdx.x == 0) partials[blockIdx.x * 32 + c] = red[0];
    __syncthreads();
    red[threadIdx.x] = h[c] * h[c];
    __syncthreads();
    for (int st = 128; st > 0; st >>= 1) {
      if (threadIdx.x < st) red[threadIdx.x] += red[threadIdx.x + st];
      __syncthreads();
    }
    if (threadIdx.x == 0) partials[blockIdx.x * 32 + 16 + c] = red[0];
    __syncthreads();
  }
}

__global__ void sn_finalize_kernel(const float* __restrict__ partials,
                                   float* __restrict__ stats) {
  const int c = threadIdx.x;
  if (c >= Hc) return;
  float s = 0.f, q = 0.f;
  for (int p = 0; p < 640; ++p) { s += partials[p * 32 + c]; q += partials[p * 32 + 16 + c]; }
  const float cnt = (float)(Bc * Nc * Kc);
  const float mean = s / cnt;
  const float var  = q / cnt - mean * mean;
  stats[c]      = mean;
  stats[16 + c] = rsqrtf(var + EPSc);
}

__global__ __launch_bounds__(256) void score_kernel(const float* __restrict__ coords,
                                                    const int* __restrict__ idx,
                                                    const float* __restrict__ sn_w1,
                                                    const float* __restrict__ sn_g1,
                                                    const float* __restrict__ sn_b1,
                                                    const float* __restrict__ sn_w2,
                                                    const float* __restrict__ sn_bias2,
                                                    const float* __restrict__ stats,
                                                    float* __restrict__ score) {
  const int gid = blockIdx.x * 256 + threadIdx.x;
  const int b = gid / (Nc * Kc);
  const int rem = gid % (Nc * Kc);
  const int n = rem / Kc;
  const int j = idx[gid];
  float h[Hc];
  sn_hidden(coords + (size_t)b * Cc * Nc, n, j, sn_w1, h);
#pragma unroll
  for (int c = 0; c < Hc; ++c) {
    float hn = (h[c] - stats[c]) * stats[16 + c] * sn_g1[c] + sn_b1[c];
    h[c] = fmaxf(hn, 0.f);                                  // BN + ReLU
  }
  float s[Mc];
#pragma unroll
  for (int m = 0; m < Mc; ++m) {
    float a = sn_bias2[m];
#pragma unroll
    for (int c = 0; c < Hc; ++c) a += h[c] * sn_w2[m * Hc + c];
    s[m] = a;
  }
  float mx = s[0];
#pragma unroll
  for (int m = 1; m < Mc; ++m) mx = fmaxf(mx, s[m]);
  float sum = 0.f;
#pragma unroll
  for (int m = 0; m < Mc; ++m) { s[m] = __expf(s[m] - mx); sum += s[m]; }
  const float inv = 1.0f / sum;
#pragma unroll
  for (int m = 0; m < Mc; ++m) score[(size_t)gid * Mc + m] = s[m] * inv + BIASc;
}

// ---------------------------------------------------------------------------
// Kernel 4: assemble.  One workgroup per (b,n); thread = output channel o.
// out[b,o,n] = sum_k sum_m score * point[idx_k] - (sum_k score) * center
// ---------------------------------------------------------------------------
__global__ __launch_bounds__(64) void assemble_kernel(const float* __restrict__ features,
                                                      const float* __restrict__ matrice1,
                                                      const float* __restrict__ point,
                                                      const float* __restrict__ score,
                                                      const int* __restrict__ idx,
                                                      float* __restrict__ out) {
  const int bn = blockIdx.x;
  const int b = bn >> 10;
  const int n = bn & 1023;
  const int o = threadIdx.x;
  const float f0 = features[(b * Cc + 0) * Nc + n];
  const float f1 = features[(b * Cc + 1) * Nc + n];
  const float f2 = features[(b * Cc + 2) * Nc + n];
  const float* scr = score + (size_t)bn * Kc * Mc;
  const int* irow = idx + (size_t)bn * Kc;

  float acc = 0.f;
  float ssum[Mc];
#pragma unroll
  for (int m = 0; m < Mc; ++m) ssum[m] = 0.f;

  for (int k = 0; k < Kc; ++k) {
    const int j = irow[k];
    const float* prow = point + ((size_t)(b * Nc + j)) * 512;
    if (k + 1 < Kc)   // CDNA5: global_prefetch_b8 on next gathered row
      __builtin_prefetch(point + ((size_t)(b * Nc + irow[k + 1])) * 512 + o, 0, 1);
#pragma unroll
    for (int m = 0; m < Mc; ++m) {
      const float sc = scr[k * Mc + m];
      acc += sc * prow[m * Oc + o];
      ssum[m] += sc;
    }
  }
#pragma unroll
  for (int m = 0; m < Mc; ++m) {
    const int col = m * Oc + o;
    const float cen = f0 * matrice1[col] + f1 * matrice1[512 + col] + f2 * matrice1[1024 + col];
    acc -= ssum[m] * cen;
  }
  out[((size_t)(b * Oc + o)) * Nc + n] = acc;
}

// ---------------------------------------------------------------------------
// Kernel 5: output BN (per-O training stats over B,N) + ReLU
// ---------------------------------------------------------------------------
__global__ __launch_bounds__(256) void bnstats_kernel(const float* __restrict__ out,
                                                      float* __restrict__ bnstat) {
  const int o = blockIdx.x;
  __shared__ float rs[256], rq[256];
  float s = 0.f, q = 0.f;
  for (int i = threadIdx.x; i < Bc * Nc; i += 256) {
    const int b = i >> 10, n = i & 1023;
    const float v = out[((size_t)(b * Oc + o)) * Nc + n];
    s += v; q += v * v;
  }
  rs[threadIdx.x] = s; rq[threadIdx.x] = q;
  __syncthreads();
  for (int st = 128; st > 0; st >>= 1) {
    if (threadIdx.x < st) { rs[threadIdx.x] += rs[threadIdx.x + st]; rq[threadIdx.x] += rq[threadIdx.x + st]; }
    __syncthreads();
  }
  if (threadIdx.x == 0) {
    const float cnt = (float)(Bc * Nc);
    const float mean = rs[0] / cnt;
    const float var  = rq[0] / cnt - mean * mean;
    bnstat[o]      = mean;
    bnstat[64 + o] = rsqrtf(var + EPSc);
  }
}

__global__ __launch_bounds__(256) void bn_apply_kernel(float* __restrict__ out,
                                                       const float* __restrict__ bnstat,
                                                       const float* __restrict__ g,
                                                       const float* __restrict__ beta) {
  const int i = blockIdx.x * 256 + threadIdx.x;          // 524288 total, exact
  const int o = (i >> 10) & 63;
  float v = out[i];
  v = (v - bnstat[o]) * bnstat[64 + o] * g[o] + beta[o];
  out[i] = fmaxf(v, 0.f);
}

// ---------------------------------------------------------------------------
extern "C" void kernel_launch(void* const* d_in, const int* in_sizes, int n_in,
                              void* d_out, int out_size, void* d_ws, size_t ws_size,
                              hipStream_t stream) {
  const float* features = (const float*)d_in[0];
  const float* coords   = (const float*)d_in[1];
  const float* matrice1 = (const float*)d_in[2];
  const float* sn_w1    = (const float*)d_in[3];
  const float* sn_g1    = (const float*)d_in[4];
  const float* sn_b1    = (const float*)d_in[5];
  const float* sn_w2    = (const float*)d_in[6];
  const float* sn_bias2 = (const float*)d_in[7];
  const float* bn_g     = (const float*)d_in[8];
  const float* bn_b     = (const float*)d_in[9];
  float* out = (float*)d_out;
  char* ws = (char*)d_ws;

  // workspace layout (bytes)
  int*   idx    = (int*)  (ws + 0);          // 8*1024*20*4        =   655360
  float* point  = (float*)(ws + 655360);     // 8*1024*512*4       = 16777216
  float* score  = (float*)(ws + 17432576);   // 8*1024*20*8*4      =  5242880
  float* part   = (float*)(ws + 22675456);   // 640*32*4           =    81920
  float* snstat = (float*)(ws + 22757376);   // 32*4
  float* bnstat = (float*)(ws + 22757504);   // 128*4

  knn_kernel        <<<Bc * (Nc / 16), 512, 0, stream>>>(coords, idx);
  point_kernel      <<<Bc * (Nc / 16), 256, 0, stream>>>(features, matrice1, point);
  sn_partial_kernel <<<640, 256, 0, stream>>>(coords, idx, sn_w1, part);
  sn_finalize_kernel<<<1, 16, 0, stream>>>(part, snstat);
  score_kernel      <<<640, 256, 0, stream>>>(coords, idx, sn_w1, sn_g1, sn_b1,
                                              sn_w2, sn_bias2, snstat, score);
  assemble_kernel   <<<Bc * Nc, 64, 0, stream>>>(features, matrice1, point, score, idx, out);
  bnstats_kernel    <<<Oc, 256, 0, stream>>>(out, bnstat);
  bn_apply_kernel   <<<(Bc * Oc * Nc) / 256, 256, 0, stream>>>(out, bnstat, bn_g, bn_b);
}